// GAT_negbin_29832842838724
// MI455X (gfx1250) — compile-verified
//
#include <hip/hip_runtime.h>
#include <hip/hip_bf16.h>
#include <math.h>

typedef __attribute__((ext_vector_type(2))) float v2f;
typedef __attribute__((ext_vector_type(8))) float v8f;

#define NEG_SLOPE 0.2f
#define FDIM 128
#define ASTR 68  // LDS row stride (floats): 16B-aligned rows, conflict-free

// ---- monotonic float<->uint key for atomicMax on floats ----
__device__ __forceinline__ unsigned f2key(float f) {
  unsigned u = __float_as_uint(f);
  return (u & 0x80000000u) ? ~u : (u | 0x80000000u);
}
__device__ __forceinline__ float key2f(unsigned k) {
  return (k & 0x80000000u) ? __uint_as_float(k & 0x7FFFFFFFu)
                           : __uint_as_float(~k);
}

// low 32 bits of a generic pointer to LDS == group-segment byte offset
__device__ __forceinline__ unsigned lds_off32(const void* p) {
  return (unsigned)(uintptr_t)p;
}

// ============================================================
// C[N,128] = A[N,K] * W[K,128]   (f32 WMMA 16x16x4, wave32)
// block = 256 threads = 8 waves; wave w -> cols [16w,16w+16)
// A panels (16x64) double-buffered in LDS via async-to-LDS DMA.
// ============================================================
__global__ __launch_bounds__(256) void gemm_f32_wmma(
    const float* __restrict__ A, const float* __restrict__ W,
    float* __restrict__ C, int N, int K) {
  __shared__ float As[2][16 * ASTR];
  const int tid = threadIdx.x;
  const int lane = tid & 31;
  const int wave = tid >> 5;
  const int rowBase = blockIdx.x * 16;
  const int colBase = wave * 16;
  const int ln16 = lane & 15;
  const int koff = (lane >> 4) * 2;  // A/B frag K offset per half-wave

  // per-thread async copy slot: 16 bytes of the 16x64 panel
  const int lr = tid >> 4;        // panel row 0..15
  const int lc = (tid & 15) * 4;  // panel col (floats)
  const int grow = rowBase + lr;
  const bool rowOK = grow < N;

  v8f c = {0.f, 0.f, 0.f, 0.f, 0.f, 0.f, 0.f, 0.f};

  // prologue: async DMA panel 0 into buffer 0
  if (rowOK) {
    unsigned l0 = lds_off32(&As[0][lr * ASTR + lc]);
    unsigned long long g0 =
        (unsigned long long)(const void*)(A + (size_t)grow * K + lc);
    asm volatile("global_load_async_to_lds_b128 %0, %1, off" ::"v"(l0),
                 "v"(g0)
                 : "memory");
  }
  asm volatile("s_wait_asynccnt 0x0" ::: "memory");
  __syncthreads();

  int buf = 0;
  for (int k0 = 0; k0 < K; k0 += 64, buf ^= 1) {
    // kick off next panel's DMA while computing on current one
    if (k0 + 64 < K && rowOK) {
      unsigned ln = lds_off32(&As[buf ^ 1][lr * ASTR + lc]);
      unsigned long long gn = (unsigned long long)(const void*)(
          A + (size_t)grow * K + (k0 + 64) + lc);
      asm volatile("global_load_async_to_lds_b128 %0, %1, off" ::"v"(ln),
                   "v"(gn)
                   : "memory");
    }
    const float* as = &As[buf][0];
#pragma unroll
    for (int kk = 0; kk < 64; kk += 4) {
      v2f a;
      a.x = as[ln16 * ASTR + kk + koff + 0];
      a.y = as[ln16 * ASTR + kk + koff + 1];
      const float* wp = W + (size_t)(k0 + kk + koff) * FDIM + colBase + ln16;
      v2f b;
      b.x = wp[0];
      b.y = wp[FDIM];
      c = __builtin_amdgcn_wmma_f32_16x16x4_f32(
          false, a, false, b, (short)0, c, false, false);
    }
    asm volatile("s_wait_asynccnt 0x0" ::: "memory");
    __syncthreads();
  }

  const int hi8 = (lane >> 4) * 8;
#pragma unroll
  for (int v = 0; v < 8; ++v) {
    int row = rowBase + v + hi8;
    if (row < N) C[(size_t)row * FDIM + colBase + ln16] = c[v];
  }
}

// ============================================================
// per-node attention dots: asrc[n]=h[n].att_s, adst[n]=h[n].att_d
// one wave32 per node, 4 channels per lane
// ============================================================
__global__ __launch_bounds__(256) void att_dots(
    const float* __restrict__ h, const float* __restrict__ att_s,
    const float* __restrict__ att_d, float* __restrict__ asrc,
    float* __restrict__ adst, int N) {
  int node = blockIdx.x * (blockDim.x >> 5) + (threadIdx.x >> 5);
  int lane = threadIdx.x & 31;
  if (node >= N) return;
  float4 hv = *(const float4*)(h + (size_t)node * FDIM + lane * 4);
  float4 s4 = *(const float4*)(att_s + lane * 4);
  float4 d4 = *(const float4*)(att_d + lane * 4);
  float ss = hv.x * s4.x + hv.y * s4.y + hv.z * s4.z + hv.w * s4.w;
  float dd = hv.x * d4.x + hv.y * d4.y + hv.z * d4.z + hv.w * d4.w;
#pragma unroll
  for (int m = 16; m >= 1; m >>= 1) {
    ss += __shfl_xor(ss, m, 32);
    dd += __shfl_xor(dd, m, 32);
  }
  if (lane == 0) {
    asrc[node] = ss;
    adst[node] = dd;
  }
}

// ============================================================
// edge pass 1: logits + segment max (uint-key atomicMax)
// ============================================================
__global__ __launch_bounds__(256) void edge_logit_max(
    const int* __restrict__ src, const int* __restrict__ dst,
    const float* __restrict__ asrc, const float* __restrict__ adst,
    float* __restrict__ ebuf, unsigned* __restrict__ maxk, int E) {
  int e = blockIdx.x * blockDim.x + threadIdx.x;
  if (e >= E) return;
  float v = asrc[src[e]] + adst[dst[e]];
  float l = v > 0.f ? v : NEG_SLOPE * v;
  ebuf[e] = l;
  atomicMax(&maxk[dst[e]], f2key(l));
}

// edge pass 2: p = exp(logit - max), segment sum
__global__ __launch_bounds__(256) void edge_exp_sum(
    const int* __restrict__ dst, float* __restrict__ ebuf,
    const unsigned* __restrict__ maxk, float* __restrict__ denom, int E) {
  int e = blockIdx.x * blockDim.x + threadIdx.x;
  if (e >= E) return;
  int d = dst[e];
  float p = __expf(ebuf[e] - key2f(maxk[d]));
  ebuf[e] = p;
  atomicAdd(&denom[d], p);
}

// edge pass 3: acc[dst] += alpha * h[src]; wave32 per edge, 4 ch/lane
__global__ __launch_bounds__(256) void edge_aggregate(
    const int* __restrict__ src, const int* __restrict__ dst,
    const float* __restrict__ ebuf, const float* __restrict__ denom,
    const float* __restrict__ h, float* __restrict__ acc,
    float* __restrict__ alpha_out, int E) {
  int e = blockIdx.x * (blockDim.x >> 5) + (threadIdx.x >> 5);
  int lane = threadIdx.x & 31;
  if (e >= E) return;
  int s = src[e], d = dst[e];
  float alpha = ebuf[e] / fmaxf(denom[d], 1e-16f);
  if (alpha_out != nullptr && lane == 0) alpha_out[e] = alpha;
  float4 hv = *(const float4*)(h + (size_t)s * FDIM + lane * 4);
  float* o = acc + (size_t)d * FDIM + lane * 4;
  atomicAdd(o + 0, alpha * hv.x);
  atomicAdd(o + 1, alpha * hv.y);
  atomicAdd(o + 2, alpha * hv.z);
  atomicAdd(o + 3, alpha * hv.w);
}

// out = acc + bias (optional relu)
__global__ __launch_bounds__(256) void bias_act(
    const float* __restrict__ acc, const float* __restrict__ bias,
    float* __restrict__ out, int total, int do_relu) {
  int i = blockIdx.x * blockDim.x + threadIdx.x;
  if (i >= total) return;
  float v = acc[i] + bias[i & (FDIM - 1)];
  out[i] = do_relu ? fmaxf(v, 0.f) : v;
}

// ---- self-loop kernels (layer 3 only); run after edge pass peers ----
__global__ __launch_bounds__(256) void self_logit_max(
    const float* __restrict__ asrc, const float* __restrict__ adst,
    float* __restrict__ pself, unsigned* __restrict__ maxk, int N) {
  int i = blockIdx.x * blockDim.x + threadIdx.x;
  if (i >= N) return;
  float v = asrc[i] + adst[i];
  float l = v > 0.f ? v : NEG_SLOPE * v;
  pself[i] = l;
  unsigned k = f2key(l);
  if (k > maxk[i]) maxk[i] = k;  // no other writers at this point
}

__global__ __launch_bounds__(256) void self_exp_sum(
    float* __restrict__ pself, const unsigned* __restrict__ maxk,
    float* __restrict__ denom, int N) {
  int i = blockIdx.x * blockDim.x + threadIdx.x;
  if (i >= N) return;
  float p = __expf(pself[i] - key2f(maxk[i]));
  pself[i] = p;
  denom[i] += p;
}

__global__ __launch_bounds__(256) void self_aggregate(
    const float* __restrict__ pself, const float* __restrict__ denom,
    const float* __restrict__ h, float* __restrict__ acc, int N) {
  int idx = blockIdx.x * blockDim.x + threadIdx.x;
  if (idx >= N * FDIM) return;
  int i = idx >> 7;
  acc[idx] += (pself[i] / fmaxf(denom[i], 1e-16f)) * h[idx];
}

extern "C" void kernel_launch(void* const* d_in, const int* in_sizes, int n_in,
                              void* d_out, int out_size, void* d_ws,
                              size_t ws_size, hipStream_t stream) {
  (void)n_in; (void)out_size; (void)ws_size;
  const float* x   = (const float*)d_in[0];
  const int*   ei  = (const int*)d_in[1];
  const float* W1  = (const float*)d_in[2];
  const float* as1 = (const float*)d_in[3];
  const float* ad1 = (const float*)d_in[4];
  const float* b1  = (const float*)d_in[5];
  const float* W2  = (const float*)d_in[6];
  const float* as2 = (const float*)d_in[7];
  const float* ad2 = (const float*)d_in[8];
  const float* b2  = (const float*)d_in[9];
  const float* Wv  = (const float*)d_in[10];
  const float* asv = (const float*)d_in[11];
  const float* adv = (const float*)d_in[12];
  const float* bv  = (const float*)d_in[13];

  const int IN_CH = 512;
  const int N = in_sizes[0] / IN_CH;
  const int E = in_sizes[1] / 2;
  const int* src = ei;
  const int* dst = ei + E;
  const int NF = N * FDIM;

  // ---- workspace layout (f32) ----
  float*    hbuf  = (float*)d_ws;            // [N,128] GEMM output
  float*    acc   = hbuf + (size_t)NF;       // [N,128] aggregation accum
  float*    feat  = acc + (size_t)NF;        // [N,128] layer-1 activated out
  float*    asrc  = feat + (size_t)NF;       // [N]
  float*    adst  = asrc + N;                // [N]
  unsigned* maxk  = (unsigned*)(adst + N);   // [N]
  float*    denom = (float*)(maxk + N);      // [N]
  float*    pself = denom + N;               // [N]
  float*    ebuf  = pself + N;               // [E]

  float* mean   = (float*)d_out;             // [N,128]
  float* var    = mean + (size_t)NF;         // [N,128]
  float* alphao = var + (size_t)NF;          // [E]

  const dim3 B(256);
  const int gGemm  = (N + 15) / 16;
  const int gNodeW = (N + 7) / 8;       // wave-per-node
  const int gEdge  = (E + 255) / 256;
  const int gEdgeW = (E + 7) / 8;       // wave-per-edge
  const int gNF    = (NF + 255) / 256;
  const int gN     = (N + 255) / 256;

  // =================== layer 1: h1 = GAT(x; W1) , relu ===================
  gemm_f32_wmma<<<gGemm, B, 0, stream>>>(x, W1, hbuf, N, IN_CH);
  att_dots<<<gNodeW, B, 0, stream>>>(hbuf, as1, ad1, asrc, adst, N);
  hipMemsetAsync(maxk, 0, (size_t)N * 4, stream);
  hipMemsetAsync(denom, 0, (size_t)N * 4, stream);
  hipMemsetAsync(acc, 0, (size_t)NF * 4, stream);
  edge_logit_max<<<gEdge, B, 0, stream>>>(src, dst, asrc, adst, ebuf, maxk, E);
  edge_exp_sum<<<gEdge, B, 0, stream>>>(dst, ebuf, maxk, denom, E);
  edge_aggregate<<<gEdgeW, B, 0, stream>>>(src, dst, ebuf, denom, hbuf, acc,
                                           nullptr, E);
  bias_act<<<gNF, B, 0, stream>>>(acc, b1, feat, NF, 1);

  // =================== layer 2: mean, alpha = GAT(h1; W2) ================
  gemm_f32_wmma<<<gGemm, B, 0, stream>>>(feat, W2, hbuf, N, FDIM);
  att_dots<<<gNodeW, B, 0, stream>>>(hbuf, as2, ad2, asrc, adst, N);
  hipMemsetAsync(maxk, 0, (size_t)N * 4, stream);
  hipMemsetAsync(denom, 0, (size_t)N * 4, stream);
  hipMemsetAsync(acc, 0, (size_t)NF * 4, stream);
  edge_logit_max<<<gEdge, B, 0, stream>>>(src, dst, asrc, adst, ebuf, maxk, E);
  edge_exp_sum<<<gEdge, B, 0, stream>>>(dst, ebuf, maxk, denom, E);
  edge_aggregate<<<gEdgeW, B, 0, stream>>>(src, dst, ebuf, denom, hbuf, acc,
                                           alphao, E);
  bias_act<<<gNF, B, 0, stream>>>(acc, b2, mean, NF, 0);

  // ============ layer 3: var = GAT(mean; Wv) with self loops =============
  gemm_f32_wmma<<<gGemm, B, 0, stream>>>(mean, Wv, hbuf, N, FDIM);
  att_dots<<<gNodeW, B, 0, stream>>>(hbuf, asv, adv, asrc, adst, N);
  hipMemsetAsync(maxk, 0, (size_t)N * 4, stream);
  hipMemsetAsync(denom, 0, (size_t)N * 4, stream);
  hipMemsetAsync(acc, 0, (size_t)NF * 4, stream);
  edge_logit_max<<<gEdge, B, 0, stream>>>(src, dst, asrc, adst, ebuf, maxk, E);
  self_logit_max<<<gN, B, 0, stream>>>(asrc, adst, pself, maxk, N);
  edge_exp_sum<<<gEdge, B, 0, stream>>>(dst, ebuf, maxk, denom, E);
  self_exp_sum<<<gN, B, 0, stream>>>(pself, maxk, denom, N);
  edge_aggregate<<<gEdgeW, B, 0, stream>>>(src, dst, ebuf, denom, hbuf, acc,
                                           nullptr, E);
  self_aggregate<<<gNF, B, 0, stream>>>(pself, denom, hbuf, acc, N);
  bias_act<<<gNF, B, 0, stream>>>(acc, bv, var, NF, 0);
}